// BaseShiftNet_5626407158028
// MI455X (gfx1250) — compile-verified
//
#include <hip/hip_runtime.h>
#include <hip/hip_bf16.h>
#include <cstdint>

#define N_  4
#define T_  24
#define C_  4
#define H_  256
#define W_  256
#define HW_ (H_ * W_)
#define NT_ (N_ * T_)
#define EPS_ 1e-8f
#define REF_DAY_ 182

typedef float v4f __attribute__((ext_vector_type(4)));

#if defined(__HIP_DEVICE_COMPILE__)
#if __has_builtin(__builtin_amdgcn_global_load_async_to_lds_b128)
#define USE_ASYNC_LDS 1
#endif
#endif

#if defined(USE_ASYNC_LDS)
typedef int v4i_ __attribute__((ext_vector_type(4)));
typedef __attribute__((address_space(1))) v4i_* gptr_v4i;
typedef __attribute__((address_space(3))) v4i_* lptr_v4i;
#endif

// ---------------------------------------------------------------------------
// Kernel 1: per-(n,t,c) min/max over H*W for c in {0,1,2}.
// 288 blocks x 256 threads; each thread streams 256 floats (64x b128) via the
// CDNA5 async global->LDS pipe, ping-pong double-buffered so one copy is
// always in flight (s_wait_asynccnt 0x1). Then wave32 shfl reduction +
// cross-wave LDS reduction. Writes min and 1/(max-min+eps).
// ---------------------------------------------------------------------------
__global__ __launch_bounds__(256) void kmm(const float* __restrict__ x,
                                           float* __restrict__ mins,
                                           float* __restrict__ scales) {
    const int slice = blockIdx.x;       // 0..287 = (n*T+t)*3 + c
    const int nt    = slice / 3;
    const int c     = slice % 3;
    const float* src = x + ((size_t)(nt * C_ + c)) * HW_;
    const int tid = threadIdx.x;

    float vmin =  3.402823466e38f;
    float vmax = -3.402823466e38f;

#if defined(USE_ASYNC_LDS)
    __shared__ v4f stage[2][256];       // 8 KB ping-pong
    const v4f* src4 = (const v4f*)src;
    // prologue: start copy for k=0 into buffer 0
    __builtin_amdgcn_global_load_async_to_lds_b128(
        (gptr_v4i)(uintptr_t)(src4 + tid),
        (lptr_v4i)(uintptr_t)&stage[0][tid], 0, 0);
    for (int k = 0; k < 64; ++k) {
        const int cur = k & 1;
        if (k + 1 < 64) {
            // start copy k+1 into the other buffer; its previous contents
            // were consumed (VALU below) before this point in program order.
            __builtin_amdgcn_global_load_async_to_lds_b128(
                (gptr_v4i)(uintptr_t)(src4 + ((k + 1) * 256 + tid)),
                (lptr_v4i)(uintptr_t)&stage[cur ^ 1][tid], 0, 0);
            asm volatile("s_wait_asynccnt 0x1" ::: "memory"); // copy k done
        } else {
            asm volatile("s_wait_asynccnt 0x0" ::: "memory");
        }
        v4f v = stage[cur][tid];
        vmin = fminf(vmin, fminf(fminf(v.x, v.y), fminf(v.z, v.w)));
        vmax = fmaxf(vmax, fmaxf(fmaxf(v.x, v.y), fmaxf(v.z, v.w)));
    }
#else
    const v4f* src4 = (const v4f*)src;
    for (int k = 0; k < 64; ++k) {
        v4f v = src4[k * 256 + tid];
        vmin = fminf(vmin, fminf(fminf(v.x, v.y), fminf(v.z, v.w)));
        vmax = fmaxf(vmax, fmaxf(fmaxf(v.x, v.y), fmaxf(v.z, v.w)));
    }
#endif

    // wave32 reduction
    for (int off = 16; off > 0; off >>= 1) {
        vmin = fminf(vmin, __shfl_xor(vmin, off));
        vmax = fmaxf(vmax, __shfl_xor(vmax, off));
    }
    __shared__ float smin[8], smax[8];
    const int wave = tid >> 5, lane = tid & 31;
    if (lane == 0) { smin[wave] = vmin; smax[wave] = vmax; }
    __syncthreads();
    if (tid == 0) {
        float a = smin[0], b = smax[0];
        for (int w = 1; w < 8; ++w) { a = fminf(a, smin[w]); b = fmaxf(b, smax[w]); }
        mins[slice]   = a;
        scales[slice] = 1.0f / (b - a + EPS_);
    }
}

// ---------------------------------------------------------------------------
// Kernel 2: per-n argmin_t |182 - dates[n,t]| (first occurrence wins).
// ---------------------------------------------------------------------------
__global__ void ktidx(const int* __restrict__ dates, int* __restrict__ tidx) {
    const int n = threadIdx.x;
    if (n < N_) {
        int bt = 0;
        int bv = abs(REF_DAY_ - dates[n * T_]);
        for (int t = 1; t < T_; ++t) {
            int v = abs(REF_DAY_ - dates[n * T_ + t]);
            if (v < bv) { bv = v; bt = t; }
        }
        tidx[n] = bt;
    }
}

// ---------------------------------------------------------------------------
// Kernel 3: x_pairs (n,t,2,h,w). gray of normalized ch0..2 for frame t and
// for reference frame t_idx[n]. b128 loads/stores; ref-frame source rows are
// hot in L2. NT stores: the output is never re-read.
// Block = 64 threads, one per 4-pixel group of one row.
// ---------------------------------------------------------------------------
__global__ __launch_bounds__(64) void kpairs(const float* __restrict__ x,
                                             const float* __restrict__ mins,
                                             const float* __restrict__ scales,
                                             const int* __restrict__ tidx,
                                             float* __restrict__ xp) {
    const int h = blockIdx.x, t = blockIdx.y, n = blockIdx.z;
    const int j = threadIdx.x;                 // 4-pixel group
    const int nt  = n * T_ + t;
    const int ntr = n * T_ + tidx[n];
    const size_t off4 = ((size_t)h * W_) / 4 + j;   // in v4f units

    const float wgt[3] = {0.299f, 0.587f, 0.114f};
    v4f g0 = (v4f)0.f;
    v4f g1 = (v4f)0.f;
#pragma unroll
    for (int c = 0; c < 3; ++c) {
        const v4f v  = ((const v4f*)x)[((size_t)(nt  * C_ + c)) * (HW_ / 4) + off4];
        const v4f vr = ((const v4f*)x)[((size_t)(ntr * C_ + c)) * (HW_ / 4) + off4];
        const float mn0 = mins[nt * 3 + c],  sc0 = scales[nt * 3 + c];
        const float mn1 = mins[ntr * 3 + c], sc1 = scales[ntr * 3 + c];
        g0 += wgt[c] * ((v  - mn0) * sc0);
        g1 += wgt[c] * ((vr - mn1) * sc1);
    }
    v4f* o0 = (v4f*)xp + ((size_t)(nt * 2 + 0)) * (HW_ / 4) + off4;
    v4f* o1 = (v4f*)xp + ((size_t)(nt * 2 + 1)) * (HW_ / 4) + off4;
    __builtin_nontemporal_store(g0, o0);
    __builtin_nontemporal_store(g1, o1);
}

// ---------------------------------------------------------------------------
// Kernel 4: translation-only bilinear (zeros padding, align_corners=False).
// Shift is uniform per image: floor(j - tx) = j + floor(-tx); weights are
// per-image scalars. Each thread computes 4 outputs from 2 rows x 5 shared
// source columns (10 loads, 1 b128 NT store). Block = 64 threads per row.
// ---------------------------------------------------------------------------
__global__ __launch_bounds__(64) void kshift(const float* __restrict__ x,
                                             const float* __restrict__ thetas,
                                             float* __restrict__ out) {
    const int i = blockIdx.x, c = blockIdx.y, b = blockIdx.z;
    const int j = threadIdx.x;                 // 4-pixel group

    const float tx = thetas[2 * b + 0];
    const float ty = thetas[2 * b + 1];
    const float ax = -tx, ay = -ty;
    const float fx = floorf(ax), fy = floorf(ay);
    const float wx = ax - fx,  wy = ay - fy;
    const int dx = (int)fx, dy = (int)fy;

    const int x0 = 4 * j + dx;                 // leftmost source column needed
    const int y0 = i + dy, y1 = y0 + 1;
    const bool by0 = (y0 >= 0) & (y0 < H_);
    const bool by1 = (y1 >= 0) & (y1 < H_);

    const float* img = x + ((size_t)(b * C_ + c)) * HW_;
    const float* row0 = img + (size_t)y0 * W_;
    const float* row1 = img + (size_t)y1 * W_;

    float r0[5], r1[5];
#pragma unroll
    for (int q = 0; q < 5; ++q) {
        const int xc = x0 + q;
        const bool bx = (xc >= 0) & (xc < W_);
        r0[q] = (by0 && bx) ? row0[xc] : 0.f;
        r1[q] = (by1 && bx) ? row1[xc] : 0.f;
    }

    const float w00 = (1.f - wy) * (1.f - wx);
    const float w01 = (1.f - wy) * wx;
    const float w10 = wy * (1.f - wx);
    const float w11 = wy * wx;

    v4f o;
    o.x = r0[0] * w00 + r0[1] * w01 + r1[0] * w10 + r1[1] * w11;
    o.y = r0[1] * w00 + r0[2] * w01 + r1[1] * w10 + r1[2] * w11;
    o.z = r0[2] * w00 + r0[3] * w01 + r1[2] * w10 + r1[3] * w11;
    o.w = r0[3] * w00 + r0[4] * w01 + r1[3] * w10 + r1[4] * w11;

    v4f* dst = (v4f*)(out + ((size_t)(b * C_ + c)) * HW_ + (size_t)i * W_) + j;
    __builtin_nontemporal_store(o, dst);
}

// ---------------------------------------------------------------------------
extern "C" void kernel_launch(void* const* d_in, const int* in_sizes, int n_in,
                              void* d_out, int out_size, void* d_ws, size_t ws_size,
                              hipStream_t stream) {
    const float* x      = (const float*)d_in[0];   // (4,24,4,256,256) f32
    const int*   dates  = (const int*)d_in[1];     // (4,24) i32
    const float* thetas = (const float*)d_in[2];   // (96,2) f32

    float* out    = (float*)d_out;                       // 25,165,824 floats
    float* xpairs = out + (size_t)NT_ * C_ * HW_;        // 12,582,912 floats

    float* wsf    = (float*)d_ws;
    float* mins   = wsf;               // 288
    float* scales = wsf + 288;         // 288
    int*   tidx   = (int*)(wsf + 576); // 4

    kmm   <<<dim3(NT_ * 3), 256, 0, stream>>>(x, mins, scales);
    ktidx <<<1, 32, 0, stream>>>(dates, tidx);
    kpairs<<<dim3(H_, T_, N_),  64, 0, stream>>>(x, mins, scales, tidx, xpairs);
    kshift<<<dim3(H_, C_, NT_), 64, 0, stream>>>(x, thetas, out);
}